// VQACircuit2_3882650437015
// MI455X (gfx1250) — compile-verified
//
#include <hip/hip_runtime.h>

#define NQ 7
#define DIM 128
#define LDP 130   // LDS row stride (floats) for U in build kernel (pad vs 64-bank conflicts)
#define ASP 136   // LDS row stride (floats) for A in batch kernel (halves -> disjoint banks)

typedef __attribute__((ext_vector_type(2))) float v2f;
typedef __attribute__((ext_vector_type(8))) float v8f;

__device__ __forceinline__ int insert_zero_bit(int x, int pos) {
    int lo = (1 << pos) - 1;
    return ((x & ~lo) << 1) | (x & lo);
}

// ---------------------------------------------------------------------------
// Kernel 1: build A = Re(U^H D U), one workgroup, U simulated in LDS.
// Qubit w corresponds to bit (6 - w) of the amplitude index (MSB = qubit 0).
// ---------------------------------------------------------------------------
__global__ __launch_bounds__(256) void vqa_build_A(const float* __restrict__ w,
                                                   float* __restrict__ A_out) {
    __shared__ float Ur[DIM * LDP];
    __shared__ float Ui[DIM * LDP];
    const int tid = threadIdx.x;

    // U = I
    for (int idx = tid; idx < DIM * DIM; idx += 256) {
        int k = idx >> 7, j = idx & 127;
        Ur[k * LDP + j] = (k == j) ? 1.0f : 0.0f;
        Ui[k * LDP + j] = 0.0f;
    }
    __syncthreads();

    for (int l = 0; l < 4; ++l) {
        // --- Rot(phi, theta, omega) on every wire ---
        for (int q = 0; q < NQ; ++q) {
            float phi   = w[(l * NQ + q) * 3 + 0];
            float theta = w[(l * NQ + q) * 3 + 1];
            float omega = w[(l * NQ + q) * 3 + 2];
            float ch = cosf(0.5f * theta), sh = sinf(0.5f * theta);
            float apo = 0.5f * (phi + omega), amo = 0.5f * (phi - omega);
            float c1 = cosf(apo), s1 = sinf(apo);
            float c2 = cosf(amo), s2 = sinf(amo);
            // a = e^{-i*apo} ch ; b = -e^{+i*amo} sh ; c = e^{-i*amo} sh ; d = e^{+i*apo} ch
            float ar =  c1 * ch, ai = -s1 * ch;
            float br = -c2 * sh, bi = -s2 * sh;
            float cr =  c2 * sh, ci = -s2 * sh;
            float dr =  c1 * ch, di =  s1 * ch;
            int bit = 1 << (6 - q);
            for (int t = tid; t < 64 * DIM; t += 256) {
                int col = t & 127;
                int p   = t >> 7;
                int i0  = insert_zero_bit(p, 6 - q);
                int i1  = i0 | bit;
                float x0r = Ur[i0 * LDP + col], x0i = Ui[i0 * LDP + col];
                float x1r = Ur[i1 * LDP + col], x1i = Ui[i1 * LDP + col];
                float n0r = ar * x0r - ai * x0i + br * x1r - bi * x1i;
                float n0i = ar * x0i + ai * x0r + br * x1i + bi * x1r;
                float n1r = cr * x0r - ci * x0i + dr * x1r - di * x1i;
                float n1i = cr * x0i + ci * x0r + dr * x1i + di * x1r;
                Ur[i0 * LDP + col] = n0r; Ui[i0 * LDP + col] = n0i;
                Ur[i1 * LDP + col] = n1r; Ui[i1 * LDP + col] = n1i;
            }
            __syncthreads();
        }
        // --- CNOT ring, range r = (l % 6) + 1 ---
        int rng = (l % (NQ - 1)) + 1;
        for (int q = 0; q < NQ; ++q) {
            int ctrl = q, tgt = (q + rng) % NQ;
            int cpos = 6 - ctrl, tpos = 6 - tgt;
            int cbit = 1 << cpos, tbit = 1 << tpos;
            int plo = cpos < tpos ? cpos : tpos;
            int phi_ = cpos < tpos ? tpos : cpos;
            for (int t = tid; t < 32 * DIM; t += 256) {
                int col = t & 127;
                int p   = t >> 7;
                int i   = insert_zero_bit(insert_zero_bit(p, plo), phi_);
                int i0  = i | cbit;        // ctrl=1, tgt=0
                int i1  = i0 | tbit;       // ctrl=1, tgt=1
                float tr = Ur[i0 * LDP + col], ti = Ui[i0 * LDP + col];
                Ur[i0 * LDP + col] = Ur[i1 * LDP + col];
                Ui[i0 * LDP + col] = Ui[i1 * LDP + col];
                Ur[i1 * LDP + col] = tr;
                Ui[i1 * LDP + col] = ti;
            }
            __syncthreads();
        }
    }

    // A[i][j] = sum_k sign(k) * (Ur[k][i]Ur[k][j] + Ui[k][i]Ui[k][j])
    // sign(k) = (-1)^( MSB(k) ^ LSB(k) )   (Z on qubit0 (bit6) and qubit6 (bit0))
    for (int idx = tid; idx < DIM * DIM; idx += 256) {
        int i = idx >> 7, j = idx & 127;
        float acc = 0.0f;
        for (int k = 0; k < DIM; ++k) {
            float s = (((k >> 6) ^ k) & 1) ? -1.0f : 1.0f;
            acc += s * (Ur[k * LDP + i] * Ur[k * LDP + j] +
                        Ui[k * LDP + i] * Ui[k * LDP + j]);
        }
        A_out[idx] = acc;
    }
}

// ---------------------------------------------------------------------------
// Kernel 2: out[n] = (x_n^T A x_n) / (x_n^T x_n) via f32 WMMA (16x16x4).
// 8 waves/block, 16 batch rows per wave, 128 rows per block.
// ---------------------------------------------------------------------------
__global__ __launch_bounds__(256) void vqa_batch(const float* __restrict__ X,
                                                 const float* __restrict__ A_g,
                                                 float* __restrict__ out) {
    __shared__ float As[DIM * ASP];
    const int tid = threadIdx.x;
    for (int idx = tid; idx < DIM * DIM; idx += 256) {
        As[(idx >> 7) * ASP + (idx & 127)] = A_g[idx];
    }
    __syncthreads();

    const int wave = tid >> 5;
    const int lane = tid & 31;
    const int half = lane >> 4;    // 0: lanes 0-15, 1: lanes 16-31
    const int l16  = lane & 15;
    const long row_base = (long)blockIdx.x * 128 + (long)wave * 16;

    // Load X as WMMA A-fragments: frag kk covers K = 4kk..4kk+3.
    // 32-bit A layout: lane m = l&15; V0 holds K = 4kk + 2*half, V1 = +1.
    v2f xf[32];
    const float* xrow = X + (row_base + l16) * DIM;
    #pragma unroll
    for (int kk = 0; kk < 32; ++kk) {
        xf[kk] = *(const v2f*)(xrow + 4 * kk + 2 * half);
    }

    float dacc[8], nacc[8];
    #pragma unroll
    for (int r = 0; r < 8; ++r) { dacc[r] = 0.0f; nacc[r] = 0.0f; }

    for (int nt = 0; nt < 8; ++nt) {
        v8f acc = {};
        #pragma unroll
        for (int kk = 0; kk < 32; ++kk) {
            int k0 = 4 * kk + 2 * half;      // B layout mirrors A: halves offset by K+2
            v2f b;
            b[0] = As[(k0    ) * ASP + nt * 16 + l16];
            b[1] = As[(k0 + 1) * ASP + nt * 16 + l16];
            acc = __builtin_amdgcn_wmma_f32_16x16x4_f32(
                false, xf[kk], false, b, (short)0, acc, false, false);
        }
        // C layout: VGPR r, this lane -> M = half*8 + r, N = nt*16 + l16
        #pragma unroll
        for (int r = 0; r < 8; ++r) {
            float xv = X[(row_base + half * 8 + r) * DIM + nt * 16 + l16];
            dacc[r] += acc[r] * xv;
            nacc[r] += xv * xv;
        }
    }

    // Reduce across the 16 lanes of each half (xor masks 1,2,4,8 stay in-half).
    #pragma unroll
    for (int r = 0; r < 8; ++r) {
        float d = dacc[r], n = nacc[r];
        #pragma unroll
        for (int m = 1; m < 16; m <<= 1) {
            d += __shfl_xor(d, m, 32);
            n += __shfl_xor(n, m, 32);
        }
        if (l16 == 0) {
            out[row_base + half * 8 + r] = d / n;
        }
    }
}

extern "C" void kernel_launch(void* const* d_in, const int* in_sizes, int n_in,
                              void* d_out, int out_size, void* d_ws, size_t ws_size,
                              hipStream_t stream) {
    const float* x = (const float*)d_in[0];   // (BATCH, 128) f32
    const float* w = (const float*)d_in[1];   // (4, 7, 3) f32
    float* out = (float*)d_out;               // (BATCH,) f32
    float* A   = (float*)d_ws;                // 128*128 f32 scratch

    vqa_build_A<<<1, 256, 0, stream>>>(w, A);

    int rows = in_sizes[0] / DIM;             // 262144
    int nblocks = rows / 128;                 // 2048 (16 rows/wave * 8 waves)
    vqa_batch<<<nblocks, 256, 0, stream>>>(x, A, out);
}